// PointPillarsScatter_48911087567669
// MI455X (gfx1250) — compile-verified
//
#include <hip/hip_runtime.h>

// PointPillars scatter for MI455X (gfx1250).
// out canvas: (B=4, C=64, NY=496, NX=432) float32, ~219.4 MB.
//
// Roofline: pure data movement, 0 FLOPs. ~219 MB zero-fill stores dominate
// (~9.4 us at 23.3 TB/s HBM); scatter adds ~16 MB coalesced b128 reads and
// 4.1M scattered 4B stores whose target lines are L2-resident (192 MB L2 ~
// canvas size) and already dirty from the zero-fill, so they merge without
// extra HBM round trips. WMMA/TDM are intentionally not used: a 0-FLOP
// permutation gains nothing from matrix units, and per-voxel TDM descriptors
// (64,000 x 256 B, SGPR-serialized setup) are far below VMEM-path throughput.

#define PP_NY 496
#define PP_NX 432
#define PP_C 64

// ---------------------------------------------------------------------------
// Kernel 1: zero the canvas with 128-bit stores (global_store_b128).
// One float4 per thread, lane-dense: each wave32 store covers a contiguous
// 512 B. Default (RT) temporal hint on purpose: keep zeroed lines in L2 so
// the subsequent scattered 4B writes hit dirty resident lines.
// Grid: 54,853,632 / 4 = 13,713,408 float4 / 256 = 53,568 blocks exact.
// ---------------------------------------------------------------------------
__global__ void pp_zero_canvas(float4* __restrict__ out, int n4) {
    int i = blockIdx.x * blockDim.x + threadIdx.x;
    if (i < n4) {
        float4 z;
        z.x = 0.0f; z.y = 0.0f; z.z = 0.0f; z.w = 0.0f;
        out[i] = z;  // -> global_store_b128
    }
}

// ---------------------------------------------------------------------------
// Kernel 2: scatter. Thread t handles voxel v = t>>4 and 4 channels
// c = 4*(t&15) .. 4*(t&15)+3.
//   * Feature read: one float4 (global_load_b128); 16 consecutive lanes cover
//     a voxel's full 256 B row -> fully coalesced.
//   * Coord reads: 2 distinct voxels per wave32 -> same-address coalescing,
//     1-2 transactions per field.
//   * Stores: 4 scattered b32 per thread (channel stride NY*NX*4 = 857 KB) --
//     unavoidable for the (B,C,NY,NX) output layout; absorbed by L2.
// Grid: 64000 voxels * 16 threads = 1,024,000 / 256 = 4,000 blocks exact.
// ---------------------------------------------------------------------------
__global__ void pp_scatter(const float4* __restrict__ vf4,
                           const int* __restrict__ coors,
                           float* __restrict__ out,
                           int n_voxels) {
    int t  = blockIdx.x * blockDim.x + threadIdx.x;
    int v  = t >> 4;          // voxel index
    int cq = t & 15;          // which float4 of the 64-channel row
    if (v >= n_voxels) return;

    // coors row: [b, z, y, x]
    int b = coors[v * 4 + 0];
    int y = coors[v * 4 + 2];
    int x = coors[v * 4 + 3];

    float4 f = vf4[v * (PP_C / 4) + cq];   // global_load_b128, coalesced

    // (B, C, NY, NX) layout; base channel c0 = 4*cq
    int c0   = cq << 2;
    int idx  = ((b * PP_C + c0) * PP_NY + y) * PP_NX + x;
    const int cs = PP_NY * PP_NX;          // channel stride in elements

    out[idx]          = f.x;
    out[idx + cs]     = f.y;
    out[idx + 2 * cs] = f.z;
    out[idx + 3 * cs] = f.w;
}

extern "C" void kernel_launch(void* const* d_in, const int* in_sizes, int n_in,
                              void* d_out, int out_size, void* d_ws, size_t ws_size,
                              hipStream_t stream) {
    (void)n_in; (void)d_ws; (void)ws_size;

    const float* vf    = (const float*)d_in[0];  // (N_VOXELS, 64) f32
    const int*   coors = (const int*)d_in[1];    // (N_VOXELS, 4)  i32
    // d_in[2] = batch_size scalar (fixed 4; canvas extent comes from out_size)

    float* out = (float*)d_out;

    const int n_voxels = in_sizes[1] / 4;        // 64000
    const int n4       = out_size / 4;           // 13,713,408 float4 slots

    // Pass 1: zero the canvas (stream order sequences it before the scatter).
    {
        const int threads = 256;
        const int blocks  = (n4 + threads - 1) / threads;   // 53,568 exact
        pp_zero_canvas<<<blocks, threads, 0, stream>>>((float4*)out, n4);
    }

    // Pass 2: scatter features, 16 threads (4 channels each) per voxel.
    {
        const int threads = 256;
        const int total   = n_voxels * (PP_C / 4);           // 1,024,000
        const int blocks  = (total + threads - 1) / threads; // 4,000 exact
        pp_scatter<<<blocks, threads, 0, stream>>>(
            (const float4*)vf, coors, out, n_voxels);
    }
}